// SparseTransformer_32667521253646
// MI455X (gfx1250) — compile-verified
//
#include <hip/hip_runtime.h>
#include <hip/hip_bf16.h>

// ---------------------------------------------------------------------------
// SparseTransformer forward for MI455X (gfx1250, wave32).
// All GEMMs run on V_WMMA_F32_16X16X4_F32 (full fp32, matches reference math;
// compute sits far below the 23.3 TB/s memory roofline so fp32 WMMA is free).
// ---------------------------------------------------------------------------

typedef float v2f __attribute__((ext_vector_type(2)));
typedef float v8f __attribute__((ext_vector_type(8)));

#define B_DIM 8192
#define N_DIM 1024
#define A_DIM 1024
#define M_DIM 4096

// ---------------------------------------------------------------------------
// Generic fp32 WMMA GEMM.
//   BT = true : C[M,N] = A[M,K] * B[N,K]^T   (both operands K-contiguous)
//   BT = false: C[M,N] = A[M,K] * B[K,N]
// Epilogue: C = relu?(acc * scale + bias[col])
// Block = 256 threads = 8 waves, wave tile 32x32 (2x2 WMMA frags),
// block tile 64 (M) x 128 (N). All dims assumed divisible (they are).
// ---------------------------------------------------------------------------
template <bool BT>
__global__ __launch_bounds__(256) void gemm_wmma_f32(
    const float* __restrict__ A, const float* __restrict__ Bm,
    float* __restrict__ C, const float* __restrict__ bias,
    int Ndim, int K, float scale, int do_relu) {
  const int lane = threadIdx.x & 31;
  const int wave = threadIdx.x >> 5;
  const int wm = wave >> 2;  // 0..1
  const int wn = wave & 3;   // 0..3
  const int m0 = blockIdx.x * 64 + wm * 32;
  const int n0 = blockIdx.y * 128 + wn * 32;
  const int mr = lane & 15;           // row (A) / col (B) within 16
  const int kh = (lane >> 4) << 1;    // K sub-offset: 0 (lanes 0-15) or 2

  v8f acc[2][2] = {};

  for (int kk = 0; kk < K; kk += 4) {
    v2f a[2], b[2];
#pragma unroll
    for (int i = 0; i < 2; ++i)
      a[i] = *(const v2f*)(A + (size_t)(m0 + i * 16 + mr) * K + kk + kh);
#pragma unroll
    for (int j = 0; j < 2; ++j) {
      if (BT) {
        b[j] = *(const v2f*)(Bm + (size_t)(n0 + j * 16 + mr) * K + kk + kh);
      } else {
        v2f t;
        t.x = Bm[(size_t)(kk + kh) * Ndim + n0 + j * 16 + mr];
        t.y = Bm[(size_t)(kk + kh + 1) * Ndim + n0 + j * 16 + mr];
        b[j] = t;
      }
    }
#pragma unroll
    for (int i = 0; i < 2; ++i)
#pragma unroll
      for (int j = 0; j < 2; ++j)
        acc[i][j] = __builtin_amdgcn_wmma_f32_16x16x4_f32(
            false, a[i], false, b[j], (short)0, acc[i][j], false, false);
  }

  const int rhalf = (lane >> 4) * 8;  // M offset 0 or 8 within C tile
  const int cn = lane & 15;
#pragma unroll
  for (int i = 0; i < 2; ++i)
#pragma unroll
    for (int j = 0; j < 2; ++j)
#pragma unroll
      for (int r = 0; r < 8; ++r) {
        const int row = m0 + i * 16 + rhalf + r;
        const int col = n0 + j * 16 + cn;
        float v = acc[i][j][r] * scale;
        if (bias) v += bias[col];
        if (do_relu) v = fmaxf(v, 0.0f);
        C[(size_t)row * Ndim + col] = v;
      }
}

// ---------------------------------------------------------------------------
// Block reductions (256 threads)
// ---------------------------------------------------------------------------
__device__ __forceinline__ float block_reduce_sum(float v, float* red) {
  const int t = threadIdx.x;
  red[t] = v;
  __syncthreads();
  for (int s = 128; s > 0; s >>= 1) {
    if (t < s) red[t] += red[t + s];
    __syncthreads();
  }
  float r = red[0];
  __syncthreads();
  return r;
}

__device__ __forceinline__ float block_reduce_max(float v, float* red) {
  const int t = threadIdx.x;
  red[t] = v;
  __syncthreads();
  for (int s = 128; s > 0; s >>= 1) {
    if (t < s) red[t] = fmaxf(red[t], red[t + s]);
    __syncthreads();
  }
  float r = red[0];
  __syncthreads();
  return r;
}

// ---------------------------------------------------------------------------
// LayerNorm(relu(in_row)) for rows of length C == 1024 (one block per row).
// relu is idempotent, so the q path (already relu'd in GEMM epilogue) is fine.
// jnp.var is population variance (ddof=0): var = E[x^2] - E[x]^2.
// ---------------------------------------------------------------------------
__global__ __launch_bounds__(256) void ln_relu(
    const float* __restrict__ in, float* __restrict__ out,
    const float* __restrict__ g, const float* __restrict__ b, int C) {
  __shared__ float red[256];
  const int row = blockIdx.x;
  const int t = threadIdx.x;
  float4 r4 = *((const float4*)(in + (size_t)row * C) + t);
  r4.x = fmaxf(r4.x, 0.0f);
  r4.y = fmaxf(r4.y, 0.0f);
  r4.z = fmaxf(r4.z, 0.0f);
  r4.w = fmaxf(r4.w, 0.0f);
  float s = r4.x + r4.y + r4.z + r4.w;
  float sq = r4.x * r4.x + r4.y * r4.y + r4.z * r4.z + r4.w * r4.w;
  const float inv_c = 1.0f / (float)C;
  const float mu = block_reduce_sum(s, red) * inv_c;
  const float var = block_reduce_sum(sq, red) * inv_c - mu * mu;
  const float inv = rsqrtf(var + 1e-5f);
  const int c = t * 4;
  float4 o;
  o.x = (r4.x - mu) * inv * g[c + 0] + b[c + 0];
  o.y = (r4.y - mu) * inv * g[c + 1] + b[c + 1];
  o.z = (r4.z - mu) * inv * g[c + 2] + b[c + 2];
  o.w = (r4.w - mu) * inv * g[c + 3] + b[c + 3];
  *((float4*)(out + (size_t)row * C) + t) = o;
}

// ---------------------------------------------------------------------------
// Sparsemax over rows of length M_DIM==4096, in place (scores -> f).
// tau solves sum(relu(s - tau)) == 1, tau in [smax-1, smax]; g(tau) is
// monotone decreasing -> 40 bisection steps reach fp32 precision.
// One block of 256 threads per row, 16 scores per thread held in registers.
// ---------------------------------------------------------------------------
__global__ __launch_bounds__(256) void sparsemax_rows(float* __restrict__ f) {
  __shared__ float red[256];
  const int row = blockIdx.x;
  const int t = threadIdx.x;
  float* base = f + (size_t)row * M_DIM + t * 16;
  float s[16];
#pragma unroll
  for (int i = 0; i < 16; ++i) {
    float v = base[i];
    s[i] = (v < -1e10f) ? 0.0f : v;  // reference's where() guard
  }
  float lmax = s[0];
#pragma unroll
  for (int i = 1; i < 16; ++i) lmax = fmaxf(lmax, s[i]);
  const float smax = block_reduce_max(lmax, red);

  float lo = smax - 1.0f, hi = smax;
  for (int it = 0; it < 40; ++it) {
    const float mid = 0.5f * (lo + hi);
    float ls = 0.0f;
#pragma unroll
    for (int i = 0; i < 16; ++i) ls += fmaxf(s[i] - mid, 0.0f);
    const float tot = block_reduce_sum(ls, red);  // uniform across block
    if (tot >= 1.0f) lo = mid; else hi = mid;
  }
  const float tau = 0.5f * (lo + hi);
#pragma unroll
  for (int i = 0; i < 16; ++i) base[i] = fmaxf(s[i] - tau, 0.0f);
}

// ---------------------------------------------------------------------------
// Launch: out = [ x | x_hat | f | v ] (fp32, flat, in return order)
// ws     = [ q (B*A) | k (M*A) ] = 48 MB
// ---------------------------------------------------------------------------
extern "C" void kernel_launch(void* const* d_in, const int* in_sizes, int n_in,
                              void* d_out, int out_size, void* d_ws,
                              size_t ws_size, hipStream_t stream) {
  (void)in_sizes; (void)n_in; (void)out_size; (void)ws_size;

  const float* x    = (const float*)d_in[0];
  const float* Wq_w = (const float*)d_in[1];
  const float* Wq_b = (const float*)d_in[2];
  const float* Wk   = (const float*)d_in[3];
  const float* Wv   = (const float*)d_in[4];
  const float* gq   = (const float*)d_in[5];
  const float* bq   = (const float*)d_in[6];
  const float* gk   = (const float*)d_in[7];
  const float* bk   = (const float*)d_in[8];
  const float* gv   = (const float*)d_in[9];
  const float* bv   = (const float*)d_in[10];

  float* out      = (float*)d_out;
  float* out_x    = out;
  float* out_xhat = out_x + (size_t)B_DIM * N_DIM;
  float* out_f    = out_xhat + (size_t)B_DIM * N_DIM;
  float* out_v    = out_f + (size_t)B_DIM * M_DIM;

  float* ws_q = (float*)d_ws;                       // B_DIM * A_DIM
  float* ws_k = ws_q + (size_t)B_DIM * A_DIM;       // M_DIM * A_DIM

  // out[0] = x (pass-through)
  hipMemcpyAsync(out_x, x, sizeof(float) * (size_t)B_DIM * N_DIM,
                 hipMemcpyDeviceToDevice, stream);

  const dim3 blk(256);

  // q_pre = relu(x @ Wq^T + b)   [B, A]
  gemm_wmma_f32<true><<<dim3(B_DIM / 64, A_DIM / 128), blk, 0, stream>>>(
      x, Wq_w, ws_q, Wq_b, A_DIM, N_DIM, 1.0f, 1);
  // q = LN(q_pre), in place (relu re-applied, idempotent)
  ln_relu<<<B_DIM, blk, 0, stream>>>(ws_q, ws_q, gq, bq, A_DIM);
  // k = LN(relu(Wk))  [M, A]
  ln_relu<<<M_DIM, blk, 0, stream>>>(Wk, ws_k, gk, bk, A_DIM);
  // v = LN(relu(Wv))  [M, N] -> directly into output slot (also GEMM input)
  ln_relu<<<M_DIM, blk, 0, stream>>>(Wv, out_v, gv, bv, N_DIM);

  // scores = q @ k^T / sqrt(A)  -> written into f's output slot
  gemm_wmma_f32<true><<<dim3(B_DIM / 64, M_DIM / 128), blk, 0, stream>>>(
      ws_q, ws_k, out_f, nullptr, M_DIM, A_DIM, 0.03125f, 0);

  // f = sparsemax(scores), in place
  sparsemax_rows<<<B_DIM, blk, 0, stream>>>(out_f);

  // x_hat = f @ v   [B, N]
  gemm_wmma_f32<false><<<dim3(B_DIM / 64, N_DIM / 128), blk, 0, stream>>>(
      out_f, out_v, out_xhat, nullptr, N_DIM, M_DIM, 1.0f, 0);
}